// NeRF_net_58514634441014
// MI455X (gfx1250) — compile-verified
//
#include <hip/hip_runtime.h>

// ---------------------------------------------------------------------------
// NeRF chain as a parallel scan over rigid transforms (CDNA5 / gfx1250).
//   T_k = (R(rtp_k), D2(rtp_k)) depends only on element k  -> data parallel
//   F_{k+1} = F_k o T_k ;  out_k = trans(F0 o T0 o ... o T_k)
// Single 1024-thread workgroup: TDM-stage input to LDS, register chains,
// wave32 shuffle scan, LDS cross-wave scan, WMMA f32 16x16x4 affine apply.
// ---------------------------------------------------------------------------

#define N_PTS      6000
#define NELEM      (N_PTS * 3)          // 18000 f32 = 72000 bytes
#define BLOCK      1024
#define PER_THREAD 6
#define NSLOT      (BLOCK * PER_THREAD) // 6144 (tail = identity transforms)
#define WAVE       32
#define NWAVE      (BLOCK / WAVE)
#define NBATCH     (N_PTS / 16)         // 375 exactly

typedef unsigned int       u32;
typedef unsigned long long u64;
typedef u32   u32x4 __attribute__((ext_vector_type(4)));
typedef u32   u32x8 __attribute__((ext_vector_type(8)));
typedef float v2f   __attribute__((ext_vector_type(2)));
typedef float v8f   __attribute__((ext_vector_type(8)));

struct Pose { float m[9]; float t[3]; };   // row-major 3x3 rotation + translation

__device__ __forceinline__ Pose pose_identity() {
  Pose p;
#pragma unroll
  for (int i = 0; i < 9; ++i) p.m[i] = 0.f;
  p.m[0] = p.m[4] = p.m[8] = 1.f;
  p.t[0] = p.t[1] = p.t[2] = 0.f;
  return p;
}

// (Ma,ta) o (Mb,tb) = (Ma*Mb, Ma*tb + ta)
__device__ __forceinline__ Pose pose_compose(const Pose& a, const Pose& b) {
  Pose r;
#pragma unroll
  for (int i = 0; i < 3; ++i) {
#pragma unroll
    for (int j = 0; j < 3; ++j)
      r.m[i*3+j] = a.m[i*3+0]*b.m[0+j] + a.m[i*3+1]*b.m[3+j] + a.m[i*3+2]*b.m[6+j];
    r.t[i] = a.m[i*3+0]*b.t[0] + a.m[i*3+1]*b.t[1] + a.m[i*3+2]*b.t[2] + a.t[i];
  }
  return r;
}

__device__ __forceinline__ Pose pose_shfl_up(const Pose& p, int d) {
  Pose r;
#pragma unroll
  for (int i = 0; i < 9; ++i) r.m[i] = __shfl_up(p.m[i], d, WAVE);
#pragma unroll
  for (int i = 0; i < 3; ++i) r.t[i] = __shfl_up(p.t[i], d, WAVE);
  return r;
}

__global__ __launch_bounds__(BLOCK)
void nerf_scan_kernel(const float* __restrict__ mc,    // mainchain: 9 f32
                      const float* __restrict__ rtp,   // inner_tensor: 18000 f32
                      float* __restrict__ out)         // 6003 x 3 f32
{
  extern __shared__ float smem[];
  float* sh_t    = smem;                  // NSLOT*3 : rtp, later world translations
  float* sh_wave = smem + NSLOT * 3;      // NWAVE*12: per-wave scan aggregates
  float* sh_f0   = sh_wave + NWAVE * 12;  // 12      : frame-0 rows of M0 + C0

  const int tid  = (int)threadIdx.x;
  const int lane = tid & (WAVE - 1);
  const int wid  = tid >> 5;

  // ---- Phase 0: TDM DMA of inner_tensor -> LDS; overlap with frame-0 -------
  if (wid == 0) {
    const u32 lds_base = (u32)(size_t)sh_t;        // low 32 bits = LDS offset
    const u64 ga       = (u64)(size_t)rtp;
    u32x4 g0;
    g0.x = 1u;                                     // count=1 (valid descriptor)
    g0.y = lds_base;                               // lds_addr
    g0.z = (u32)ga;                                // global_addr[31:0]
    g0.w = (u32)((ga >> 32) & 0x01FFFFFFull)       // global_addr[56:32]
         | (2u << 30);                             // type = 2 ("image")
    u32x8 g1;
    g1.s0 = (2u << 16);                            // data_size = 4 bytes
    g1.s1 = ((u32)NELEM & 0xFFFFu) << 16;          // tensor_dim0[15:0]
    g1.s2 = ((u32)NELEM >> 16) | (1u << 16);       // tensor_dim0[31:16] | tensor_dim1=1
    g1.s3 = ((u32)NELEM & 0xFFFFu) << 16;          // tile_dim0 = 18000 (fits 16b)
    g1.s4 = 1u;                                    // tile_dim1 = 1
    g1.s5 = (u32)NELEM;                            // tensor_dim0_stride[31:0]
    g1.s6 = 0u;
    g1.s7 = 0u;
    asm volatile("tensor_load_to_lds %0, %1" :: "s"(g0), "s"(g1) : "memory");

    // Overlap with the DMA: pass through seed rows + build frame 0.
    if (tid < 9) out[tid] = mc[tid];
    if (tid == 0) {
      float ax=mc[0], ay=mc[1], az=mc[2];
      float bx=mc[3], by=mc[4], bz=mc[5];
      float cx=mc[6], cy=mc[7], cz=mc[8];
      float v1x=bx-ax, v1y=by-ay, v1z=bz-az;
      float v2x=cx-bx, v2y=cy-by, v2z=cz-bz;
      float iv  = rsqrtf(v2x*v2x + v2y*v2y + v2z*v2z);
      float ux=v2x*iv, uy=v2y*iv, uz=v2z*iv;                     // bc
      float wx=v1y*uz - v1z*uy, wy=v1z*ux - v1x*uz, wz=v1x*uy - v1y*ux;
      float iw  = rsqrtf(wx*wx + wy*wy + wz*wz);
      float nx=wx*iw, ny=wy*iw, nz=wz*iw;                        // n_n
      float qx=ny*uz - nz*uy, qy=nz*ux - nx*uz, qz=nx*uy - ny*ux;// cross(n,bc)
      // rows of M0 whose columns are (bc, cross(n,bc), n)
      sh_f0[0]=ux; sh_f0[1]=qx; sh_f0[2]=nx;
      sh_f0[3]=uy; sh_f0[4]=qy; sh_f0[5]=ny;
      sh_f0[6]=uz; sh_f0[7]=qz; sh_f0[8]=nz;
      sh_f0[9]=cx; sh_f0[10]=cy; sh_f0[11]=cz;
    }
    __builtin_amdgcn_s_wait_tensorcnt(0);
  }
  __syncthreads();

  // ---- Phase 1: per-element relative transforms + per-thread serial chain --
  Pose cur = pose_identity();
  float tloc[PER_THREAD][3];
  const int base = tid * PER_THREAD;
#pragma unroll
  for (int j = 0; j < PER_THREAD; ++j) {
    const int k = base + j;
    Pose T;
    if (k < N_PTS) {
      float r  = sh_t[3*k+0];
      float th = sh_t[3*k+1];
      float ph = sh_t[3*k+2];
      float st = __sinf(th), ct = __cosf(th);
      float sp = __sinf(ph), cp = __cosf(ph);
      float dx = r*ct, dy = r*cp*st, dz = r*sp*st;          // D2
      float iv = rsqrtf(dx*dx + dy*dy + dz*dz);
      float bx = dx*iv, by2 = dy*iv, bz = dz*iv;            // bc
      float i2 = rsqrtf(by2*by2 + bz*bz);
      float ny = -bz*i2, nz = by2*i2;                       // n = norm(cross(ex,bc)), n_x=0
      float c1x = ny*bz - nz*by2, c1y = nz*bx, c1z = -ny*bx;// cross(n,bc)
      T.m[0]=bx;  T.m[1]=c1x; T.m[2]=0.f;
      T.m[3]=by2; T.m[4]=c1y; T.m[5]=ny;
      T.m[6]=bz;  T.m[7]=c1z; T.m[8]=nz;
      T.t[0]=dx;  T.t[1]=dy;  T.t[2]=dz;
    } else {
      T = pose_identity();
    }
    cur = (j == 0) ? T : pose_compose(cur, T);
    tloc[j][0] = cur.t[0]; tloc[j][1] = cur.t[1]; tloc[j][2] = cur.t[2];
  }

  // ---- Phase 2: wave32 inclusive scan of thread aggregates (shuffles) ------
  Pose P = cur;
#pragma unroll
  for (int d = 1; d < WAVE; d <<= 1) {
    Pose Q = pose_shfl_up(P, d);
    if (lane >= d) P = pose_compose(Q, P);
  }

  // ---- Phase 3: cross-wave scan through LDS --------------------------------
  if (lane == WAVE - 1) {
#pragma unroll
    for (int i = 0; i < 9; ++i) sh_wave[wid*12 + i] = P.m[i];
#pragma unroll
    for (int i = 0; i < 3; ++i) sh_wave[wid*12 + 9 + i] = P.t[i];
  }
  __syncthreads();
  if (wid == 0) {
    Pose W;
#pragma unroll
    for (int i = 0; i < 9; ++i) W.m[i] = sh_wave[lane*12 + i];
#pragma unroll
    for (int i = 0; i < 3; ++i) W.t[i] = sh_wave[lane*12 + 9 + i];
#pragma unroll
    for (int d = 1; d < WAVE; d <<= 1) {
      Pose Q = pose_shfl_up(W, d);
      if (lane >= d) W = pose_compose(Q, W);
    }
#pragma unroll
    for (int i = 0; i < 9; ++i) sh_wave[lane*12 + i] = W.m[i];
#pragma unroll
    for (int i = 0; i < 3; ++i) sh_wave[lane*12 + 9 + i] = W.t[i];
  }
  __syncthreads();

  // Exclusive prefix for this thread, then per-element world translations.
  Pose Ew = pose_shfl_up(P, 1);
  if (lane == 0) Ew = pose_identity();
  Pose E;
  if (wid == 0) {
    E = Ew;
  } else {
    Pose Wp;
#pragma unroll
    for (int i = 0; i < 9; ++i) Wp.m[i] = sh_wave[(wid-1)*12 + i];
#pragma unroll
    for (int i = 0; i < 3; ++i) Wp.t[i] = sh_wave[(wid-1)*12 + 9 + i];
    E = pose_compose(Wp, Ew);
  }
#pragma unroll
  for (int j = 0; j < PER_THREAD; ++j) {
    const int slot = base + j;
#pragma unroll
    for (int i = 0; i < 3; ++i) {
      sh_t[slot*3 + i] = E.m[i*3+0]*tloc[j][0] + E.m[i*3+1]*tloc[j][1]
                       + E.m[i*3+2]*tloc[j][2] + E.t[i];
    }
  }
  __syncthreads();

  // ---- Phase 4: apply frame-0 affine transform, out = M0*t + C0 ------------
#if __has_builtin(__builtin_amdgcn_wmma_f32_16x16x4_f32)
  {
    const int msel = lane & 15;
    const int hi   = (lane >= 16);
    // A (16x4): rows 0..2 = [M0 row | C0], rows 3..15 = 0.
    // lanes 0-15: (K=0,K=1); lanes 16-31: (K=2,K=3)  [ISA 32-bit A 16x4 layout]
    v2f a;
    a.x = (msel < 3) ? (hi ? sh_f0[msel*3+2] : sh_f0[msel*3+0]) : 0.f;
    a.y = (msel < 3) ? (hi ? sh_f0[9 + msel] : sh_f0[msel*3+1]) : 0.f;
    for (int b = wid; b < NBATCH; b += NWAVE) {           // wave-uniform loop
      const int p0  = b * 16;
      const int idx = p0 + msel;
      // B (4x16): column n = [x;y;z;1] of point p0+n (mirrored A layout)
      v2f bb;
      bb.x = hi ? sh_t[idx*3 + 2] : sh_t[idx*3 + 0];
      bb.y = hi ? 1.0f            : sh_t[idx*3 + 1];
      v8f c = {0.f,0.f,0.f,0.f,0.f,0.f,0.f,0.f};
      v8f dmat = __builtin_amdgcn_wmma_f32_16x16x4_f32(
          /*neg_a=*/false, a, /*neg_b=*/false, bb,
          /*c_mod=*/(short)0, c, /*reuse_a=*/false, /*reuse_b=*/false);
      if (!hi) {                                          // D rows 0..2, N=lane
        float* o = out + (3 + idx) * 3;
        o[0] = dmat[0]; o[1] = dmat[1]; o[2] = dmat[2];
      }
    }
  }
#else
  {
    const float m0 = sh_f0[0], m1 = sh_f0[1], m2 = sh_f0[2];
    const float m3 = sh_f0[3], m4 = sh_f0[4], m5 = sh_f0[5];
    const float m6 = sh_f0[6], m7 = sh_f0[7], m8 = sh_f0[8];
    const float c0 = sh_f0[9], c1 = sh_f0[10], c2 = sh_f0[11];
    for (int k = tid; k < N_PTS; k += BLOCK) {
      float x = sh_t[3*k+0], y = sh_t[3*k+1], z = sh_t[3*k+2];
      float* o = out + (3 + k) * 3;
      o[0] = m0*x + m1*y + m2*z + c0;
      o[1] = m3*x + m4*y + m5*z + c1;
      o[2] = m6*x + m7*y + m8*z + c2;
    }
  }
#endif
}

extern "C" void kernel_launch(void* const* d_in, const int* in_sizes, int n_in,
                              void* d_out, int out_size, void* d_ws, size_t ws_size,
                              hipStream_t stream) {
  (void)in_sizes; (void)n_in; (void)out_size; (void)d_ws; (void)ws_size;
  const float* mc  = (const float*)d_in[0];   // mainchain_coord (3,3)
  const float* rtp = (const float*)d_in[1];   // inner_tensor (6000,3)
  float* out = (float*)d_out;                 // (6003,3)
  const size_t smem_bytes = (size_t)(NSLOT*3 + NWAVE*12 + 12) * sizeof(float);
  hipLaunchKernelGGL(nerf_scan_kernel, dim3(1), dim3(BLOCK), smem_bytes, stream,
                     mc, rtp, out);
}